// OntologyGNN_33320356283048
// MI455X (gfx1250) — compile-verified
//
#include <hip/hip_runtime.h>
#include <math.h>

typedef __attribute__((ext_vector_type(2))) float v2f;
typedef __attribute__((ext_vector_type(8))) float v8f;

__device__ __forceinline__ void atomAddF(float* p, float v) {
  __hip_atomic_fetch_add(p, v, __ATOMIC_RELAXED, __HIP_MEMORY_SCOPE_AGENT);
}

// ---------------- degree / normalization ----------------

__global__ void k_deg_init(float* __restrict__ deg, int n) {
  int i = blockIdx.x * blockDim.x + threadIdx.x;
  if (i < n) deg[i] = 1.0f;  // self loop
}

__global__ void k_deg_count(const int* __restrict__ dst, float* __restrict__ deg, int e) {
  int i = blockIdx.x * blockDim.x + threadIdx.x;
  if (i < e) atomAddF(&deg[dst[i]], 1.0f);
}

__global__ void k_dinv(float* __restrict__ deg, int n) {
  int i = blockIdx.x * blockDim.x + threadIdx.x;
  if (i < n) {
    float d = deg[i];
    deg[i] = d > 0.0f ? rsqrtf(d) : 0.0f;
  }
}

// ---------------- fp32 WMMA GEMM: H[N,F] = X[N,K] @ W[K,F] ----------------
// One wave computes a 16-row x 64-col strip. K stepped by 4 using
// V_WMMA_F32_16X16X4_F32. Fragment layouts per CDNA5 ISA 7.12.2:
//   A (16x4 f32, 2 VGPRs): lane half h (0/1), VGPR0 = A[M=lane&15, K=2h],
//                          VGPR1 = A[M=lane&15, K=2h+1]
//   B (4x16 f32, 2 VGPRs): VGPR0 = B[K=2h, N=lane&15], VGPR1 = B[K=2h+1, N]
//   C/D (16x16 f32, 8 VGPRs): elem r -> M = r + 8h, N = lane&15
__global__ __launch_bounds__(256) void k_gemm_wmma_f32(
    const float* __restrict__ X, const float* __restrict__ W,
    float* __restrict__ H, int N, int K, int F) {
  const int wavesPerBlock = blockDim.x >> 5;
  const int wave = blockIdx.x * wavesPerBlock + (threadIdx.x >> 5);
  const int colGroups = F >> 6;  // 64 columns per wave
  const int rowTiles = (N + 15) >> 4;
  const int rowTile = wave / colGroups;
  const int colGroup = wave % colGroups;
  if (rowTile >= rowTiles) return;  // uniform per-wave, keeps EXEC full

  const int lane = threadIdx.x & 31;
  const int half = lane >> 4;
  const int l16 = lane & 15;
  const int row = rowTile * 16 + l16;
  const int rowc = row < N ? row : N - 1;  // clamp loads on tail tile
  const int col0 = colGroup * 64;

  const float* __restrict__ xrow = X + (size_t)rowc * K;

  v8f acc0 = {}, acc1 = {}, acc2 = {}, acc3 = {};
  for (int k = 0; k < K; k += 4) {
    const int ka = k + 2 * half;
    v2f a;
    a.x = xrow[ka];
    a.y = xrow[ka + 1];
    const float* __restrict__ w0 = W + (size_t)ka * F + col0 + l16;
    const float* __restrict__ w1 = w0 + F;
    v2f b0, b1, b2, b3;
    b0.x = w0[0];  b0.y = w1[0];
    b1.x = w0[16]; b1.y = w1[16];
    b2.x = w0[32]; b2.y = w1[32];
    b3.x = w0[48]; b3.y = w1[48];
    acc0 = __builtin_amdgcn_wmma_f32_16x16x4_f32(false, a, false, b0, (short)0, acc0, false, false);
    acc1 = __builtin_amdgcn_wmma_f32_16x16x4_f32(false, a, false, b1, (short)0, acc1, false, false);
    acc2 = __builtin_amdgcn_wmma_f32_16x16x4_f32(false, a, false, b2, (short)0, acc2, false, false);
    acc3 = __builtin_amdgcn_wmma_f32_16x16x4_f32(false, a, false, b3, (short)0, acc3, false, false);
  }

  // store (divergence here is after all WMMA ops -> safe)
#pragma unroll
  for (int r = 0; r < 8; ++r) {
    const int m = rowTile * 16 + r + 8 * half;
    if (m < N) {
      float* __restrict__ hrow = H + (size_t)m * F + col0 + l16;
      hrow[0]  = acc0[r];
      hrow[16] = acc1[r];
      hrow[32] = acc2[r];
      hrow[48] = acc3[r];
    }
  }
}

// ---------------- self-loop init (+optional bias) ----------------
// out[i,c] = h[i,c] * dinv[i]^2 (+ bias[c]) ; F must be power of two
__global__ void k_selfloop_init(const float* __restrict__ h,
                                const float* __restrict__ dinv,
                                const float* __restrict__ bias,
                                float* __restrict__ out,
                                int n, int f, int addBias) {
  size_t i = (size_t)blockIdx.x * blockDim.x + threadIdx.x;
  size_t total = (size_t)n * f;
  if (i >= total) return;
  int node = (int)(i / f);
  int c = (int)(i & (f - 1));
  float di = dinv[node];
  float v = h[i] * di * di;
  if (addBias) v += bias[c];
  out[i] = v;
}

// ---------------- edge scatter: out[dst] += h[src] * dinv[src]*dinv[dst] ----
// One wave per edge; each lane moves F/32 floats via float4.
template <int F>
__global__ __launch_bounds__(256) void k_scatter(
    const int* __restrict__ src, const int* __restrict__ dst,
    const float* __restrict__ dinv, const float* __restrict__ h,
    float* __restrict__ out, int e) {
  const int wavesPerBlock = blockDim.x >> 5;
  const int eid = blockIdx.x * wavesPerBlock + (threadIdx.x >> 5);
  if (eid >= e) return;
  const int lane = threadIdx.x & 31;
  const int s = src[eid];
  const int d = dst[eid];
  const float norm = dinv[s] * dinv[d];
  const float4* __restrict__ hs =
      (const float4*)(h + (size_t)s * F) + lane;
  float* __restrict__ od = out + (size_t)d * F + lane * 4;
#pragma unroll
  for (int i = 0; i < F / 128; ++i) {
    float4 v = hs[i * 32];
    atomAddF(od + i * 128 + 0, v.x * norm);
    atomAddF(od + i * 128 + 1, v.y * norm);
    atomAddF(od + i * 128 + 2, v.z * norm);
    atomAddF(od + i * 128 + 3, v.w * norm);
  }
}

// ---------------- bias + relu (in place) ----------------
__global__ void k_bias_relu(float* __restrict__ a, const float* __restrict__ b,
                            int n, int f) {
  size_t i = (size_t)blockIdx.x * blockDim.x + threadIdx.x;
  size_t total = (size_t)n * f;
  if (i >= total) return;
  int c = (int)(i & (f - 1));
  float v = a[i] + b[c];
  a[i] = v > 0.0f ? v : 0.0f;
}

// ---------------- host launch ----------------

static inline size_t align256(size_t v) { return (v + 255) & ~(size_t)255; }

extern "C" void kernel_launch(void* const* d_in, const int* in_sizes, int n_in,
                              void* d_out, int out_size, void* d_ws, size_t ws_size,
                              hipStream_t stream) {
  const float* x  = (const float*)d_in[0];
  const int*   ei = (const int*)d_in[1];
  const float* W1 = (const float*)d_in[2];
  const float* b1 = (const float*)d_in[3];
  const float* W2 = (const float*)d_in[4];
  const float* b2 = (const float*)d_in[5];

  const int K1 = 256;                 // input_dim
  const int n  = in_sizes[0] / K1;    // 100000
  const int e  = in_sizes[1] / 2;     // 3200000
  const int F1 = in_sizes[3];         // 256 (hidden)
  const int F2 = in_sizes[5];         // 128 (out)

  const int* src = ei;
  const int* dst = ei + e;

  char* ws = (char*)d_ws;
  size_t off = 0;
  float* dinv = (float*)(ws + off); off += align256((size_t)n * 4);
  float* h1   = (float*)(ws + off); off += align256((size_t)n * F1 * 4);
  float* agg1 = (float*)(ws + off); off += align256((size_t)n * F1 * 4);
  float* h2   = h1;  // h1 dead once agg1 is built; reuse its space
  float* out  = (float*)d_out;

  const int T = 256;

  // 1) degrees -> dinv
  k_deg_init<<<(n + T - 1) / T, T, 0, stream>>>(dinv, n);
  k_deg_count<<<(e + T - 1) / T, T, 0, stream>>>(dst, dinv, e);
  k_dinv<<<(n + T - 1) / T, T, 0, stream>>>(dinv, n);

  // 2) h1 = x @ W1 (WMMA fp32)
  {
    int rowTiles = (n + 15) / 16;
    int waves = rowTiles * (F1 / 64);
    int blocks = (waves + 7) / 8;  // 8 waves / 256-thread block
    k_gemm_wmma_f32<<<blocks, T, 0, stream>>>(x, W1, h1, n, K1, F1);
  }

  // 3) agg1 = selfloop + edge scatter; then relu(agg1 + b1)
  {
    size_t total = (size_t)n * F1;
    k_selfloop_init<<<(unsigned)((total + T - 1) / T), T, 0, stream>>>(
        h1, dinv, b1, agg1, n, F1, /*addBias=*/0);
    int blocks = (e + 7) / 8;
    k_scatter<256><<<blocks, T, 0, stream>>>(src, dst, dinv, h1, agg1, e);
    k_bias_relu<<<(unsigned)((total + T - 1) / T), T, 0, stream>>>(agg1, b1, n, F1);
  }

  // 4) h2 = agg1 @ W2 (WMMA fp32)
  {
    int rowTiles = (n + 15) / 16;
    int waves = rowTiles * (F2 / 64);
    int blocks = (waves + 7) / 8;
    k_gemm_wmma_f32<<<blocks, T, 0, stream>>>(agg1, W2, h2, n, F1, F2);
  }

  // 5) out = selfloop(h2) + b2 + edge scatter
  {
    size_t total = (size_t)n * F2;
    k_selfloop_init<<<(unsigned)((total + T - 1) / T), T, 0, stream>>>(
        h2, dinv, b2, out, n, F2, /*addBias=*/1);
    int blocks = (e + 7) / 8;
    k_scatter<128><<<blocks, T, 0, stream>>>(src, dst, dinv, h2, out, e);
  }
}